// AssociativeEmbeddingLoss_1279900254921
// MI455X (gfx1250) — compile-verified
//
#include <hip/hip_runtime.h>

typedef __attribute__((ext_vector_type(2))) float v2f;
typedef __attribute__((ext_vector_type(8))) float v8f;

#define BIMG 64
#define NGT  128
#define HDIM 512
#define WDIM 512
// PULL_W = PUSH_W = 0.25, MARGIN = 1.0

// ---------------------------------------------------------------------------
// Kernel A: gather tl/br at match indices, compute me, per-image pull sum.
// One block per image, 128 threads (one per gt).
// ---------------------------------------------------------------------------
__global__ void ae_gather_pull(const float* __restrict__ pred,
                               const float* __restrict__ target,
                               const int*   __restrict__ match,
                               float* __restrict__ ws_me,
                               float* __restrict__ ws_pull) {
  const int b = blockIdx.x;
  const int g = threadIdx.x;              // 0..127
  // match layout (B, NGT, 2, 2) int32
  const int* m = match + (b * NGT + g) * 4;
  const int ty = m[0], tx = m[1], by = m[2], bx = m[3];
  const float tl = pred  [(size_t)(b * HDIM + ty) * WDIM + tx];
  const float br = target[(size_t)(b * HDIM + by) * WDIM + bx];
  const float me = 0.5f * (tl + br);
  ws_me[b * NGT + g] = me;
  const float dtl = tl - me, dbr = br - me;
  float p = dtl * dtl + dbr * dbr;

  __shared__ float red[NGT];
  red[g] = p;
  __syncthreads();
  for (int s = NGT / 2; s > 0; s >>= 1) {
    if (g < s) red[g] += red[g + s];
    __syncthreads();
  }
  if (g == 0) ws_pull[b] = red[0] / (float)NGT;
}

// ---------------------------------------------------------------------------
// Kernel B: per-image push via WMMA. d_ij = me_i - me_j computed as a
// rank-2 matmul with V_WMMA_F32_16X16X4_F32:
//   A (16x4) = [ me_i, -1, me_i, -1 ]   (replicated across K half-groups)
//   B (4x16) = [ 1 ; me_j ; 1 ; me_j ]
//   D = 2*(me_i - me_j)  -> scale 0.5 before |.|
// 128x128 pair matrix = 8x8 tiles of 16x16; 8 waves/block, one tile-row each.
// Diagonal elements come out exactly 0 -> contribute exactly MARGIN each;
// subtract N*MARGIN once instead of masking.
// ---------------------------------------------------------------------------
__global__ void ae_push_wmma(const float* __restrict__ ws_me,
                             float* __restrict__ ws_push) {
  const int b = blockIdx.x;
  __shared__ float s_me[NGT];
  __shared__ float s_red[8];
  const int t = threadIdx.x;              // 0..255 (8 waves of 32)
  if (t < NGT) s_me[t] = ws_me[b * NGT + t];
  __syncthreads();

  const int w    = t >> 5;                // wave id == tile row 0..7
  const int lane = t & 31;
  const int l15  = lane & 15;

  // A fragment: V0 = A[M][K0]/A[M][K2] = me_i[M]; V1 = A[M][K1]/A[M][K3] = -1
  v2f a;
  a.x = s_me[16 * w + l15];
  a.y = -1.0f;

  float acc = 0.0f;
  for (int tj = 0; tj < 8; ++tj) {
    // B fragment: V0 = B[K0][N]/B[K2][N] = 1; V1 = B[K1][N]/B[K3][N] = me_j[N]
    v2f bb;
    bb.x = 1.0f;
    bb.y = s_me[16 * tj + l15];
    v8f c = {};
    // 8 args: (neg_a, A, neg_b, B, c_mod, C, reuse_a, reuse_b)
    v8f d = __builtin_amdgcn_wmma_f32_16x16x4_f32(
        false, a, false, bb, (short)0, c, false, false);
#pragma unroll
    for (int r = 0; r < 8; ++r) {
      const float dv = 0.5f * d[r];               // me_i - me_j
      acc += fmaxf(0.0f, 1.0f - fabsf(dv));       // relu(MARGIN - |d|)
    }
  }

  // wave32 reduction
  for (int off = 16; off > 0; off >>= 1)
    acc += __shfl_xor(acc, off, 32);
  if (lane == 0) s_red[w] = acc;
  __syncthreads();

  if (t == 0) {
    float S = 0.0f;
    for (int i = 0; i < 8; ++i) S += s_red[i];
    S -= (float)NGT;                              // remove exact diagonal terms
    ws_push[b] = S / (float)(NGT * (NGT - 1));
  }
}

// ---------------------------------------------------------------------------
// Kernel C: final deterministic reduction over B images, write d_out[0..1].
// ---------------------------------------------------------------------------
__global__ void ae_final(const float* __restrict__ ws_pull,
                         const float* __restrict__ ws_push,
                         float* __restrict__ out) {
  __shared__ float sp[BIMG];
  __shared__ float sq[BIMG];
  const int t = threadIdx.x;              // 0..63
  sp[t] = ws_pull[t];
  sq[t] = ws_push[t];
  __syncthreads();
  for (int s = BIMG / 2; s > 0; s >>= 1) {
    if (t < s) { sp[t] += sp[t + s]; sq[t] += sq[t + s]; }
    __syncthreads();
  }
  if (t == 0) {
    out[0] = 0.25f * sp[0];   // PULL_W * sum(pulls)
    out[1] = 0.25f * sq[0];   // PUSH_W * sum(pushes)
  }
}

extern "C" void kernel_launch(void* const* d_in, const int* in_sizes, int n_in,
                              void* d_out, int out_size, void* d_ws, size_t ws_size,
                              hipStream_t stream) {
  const float* pred   = (const float*)d_in[0];
  const float* target = (const float*)d_in[1];
  const int*   match  = (const int*)d_in[2];   // JAX default: int64 demoted to int32

  float* ws      = (float*)d_ws;
  float* ws_me   = ws;                    // BIMG*NGT floats
  float* ws_pull = ws_me + BIMG * NGT;    // BIMG floats
  float* ws_push = ws_pull + BIMG;        // BIMG floats

  ae_gather_pull<<<BIMG, NGT, 0, stream>>>(pred, target, match, ws_me, ws_pull);
  ae_push_wmma  <<<BIMG, 256, 0, stream>>>(ws_me, ws_push);
  ae_final      <<<1, BIMG, 0, stream>>>(ws_pull, ws_push, (float*)d_out);
}